// Renderer_69947837383256
// MI455X (gfx1250) — compile-verified
//
#include <hip/hip_runtime.h>
#include <math.h>

#define NG   2048
#define IMG  128
#define NPIX (IMG*IMG)
#define TANF 0.4227932187381618f
#define REC_DW 12                      // floats per packed gaussian record
#define SMEM_BYTES (NG*REC_DW*4)       // 96 KB staged in LDS per workgroup

typedef __attribute__((ext_vector_type(2))) float v2f;
typedef __attribute__((ext_vector_type(8))) float v8f;
typedef unsigned int u32;
typedef u32 __attribute__((ext_vector_type(4))) v4u;
typedef int __attribute__((ext_vector_type(4))) v4i;
typedef int __attribute__((ext_vector_type(8))) v8i;

// ---------------------------------------------------------------------------
// Kernel 1: per-Gaussian preprocessing.
// Packed 12-float records: [g0,g1,g2,g3, g4,g5,opac',tz, r,g,b,0]
// power(p,g) = g0 + g1*x + g2*y + g3*x^2 + g4*y^2 + g5*x*y
// opac' = 0 for invalid gaussians (blend identity).
// ---------------------------------------------------------------------------
__global__ __launch_bounds__(256) void gs_preprocess(
    const float* __restrict__ pc,      // NG x 59
    const float* __restrict__ viewm,   // 4x4 row-major
    const float* __restrict__ projm,   // 4x4 row-major
    const float* __restrict__ campos,  // 3
    float* __restrict__ recs)          // NG x 12
{
  int g = blockIdx.x * blockDim.x + threadIdx.x;
  if (g >= NG) return;
  const float* p = pc + g * 59;

  float mx = p[0], my = p[1], mz = p[2];

  float t0 = viewm[0]*mx + viewm[1]*my + viewm[2]*mz  + viewm[3];
  float t1 = viewm[4]*mx + viewm[5]*my + viewm[6]*mz  + viewm[7];
  float tz = viewm[8]*mx + viewm[9]*my + viewm[10]*mz + viewm[11];

  float lim  = 1.3f * TANF;
  float invz = 1.0f / tz;
  float txn = fminf(fmaxf(t0*invz, -lim), lim) * tz;
  float tyn = fminf(fmaxf(t1*invz, -lim), lim) * tz;
  float fx = (float)IMG / (2.0f * TANF);
  float fy = (float)IMG / (2.0f * TANF);
  float J00 = fx*invz, J02 = -fx*txn*invz*invz;
  float J11 = fy*invz, J12 = -fy*tyn*invz*invz;

  float s3[3];
  s3[0] = __expf(p[52]); s3[1] = __expf(p[53]); s3[2] = __expf(p[54]);
  float qw = p[55], qx = p[56], qy = p[57], qz = p[58];
  float qn = rsqrtf(qw*qw + qx*qx + qy*qy + qz*qz);
  qw*=qn; qx*=qn; qy*=qn; qz*=qn;
  float R[3][3];
  R[0][0]=1.f-2.f*(qy*qy+qz*qz); R[0][1]=2.f*(qx*qy-qw*qz); R[0][2]=2.f*(qx*qz+qw*qy);
  R[1][0]=2.f*(qx*qy+qw*qz);     R[1][1]=1.f-2.f*(qx*qx+qz*qz); R[1][2]=2.f*(qy*qz-qw*qx);
  R[2][0]=2.f*(qx*qz-qw*qy);     R[2][1]=2.f*(qy*qz+qw*qx);     R[2][2]=1.f-2.f*(qx*qx+qy*qy);
  float L[3][3];
  #pragma unroll
  for (int i = 0; i < 3; ++i)
    #pragma unroll
    for (int j = 0; j < 3; ++j) L[i][j] = R[i][j]*s3[j];
  float Sig[3][3];
  #pragma unroll
  for (int i = 0; i < 3; ++i)
    #pragma unroll
    for (int k = 0; k < 3; ++k) {
      float s = 0.f;
      #pragma unroll
      for (int j = 0; j < 3; ++j) s += L[i][j]*L[k][j];
      Sig[i][k] = s;
    }

  float V[3][3] = {{viewm[0],viewm[1],viewm[2]},
                   {viewm[4],viewm[5],viewm[6]},
                   {viewm[8],viewm[9],viewm[10]}};
  float T2[2][3];
  #pragma unroll
  for (int j = 0; j < 3; ++j) {
    T2[0][j] = J00*V[0][j] + J02*V[2][j];
    T2[1][j] = J11*V[1][j] + J12*V[2][j];
  }
  float tmp[2][3];
  #pragma unroll
  for (int r = 0; r < 2; ++r)
    #pragma unroll
    for (int k = 0; k < 3; ++k) {
      float s = 0.f;
      #pragma unroll
      for (int j = 0; j < 3; ++j) s += T2[r][j]*Sig[j][k];
      tmp[r][k] = s;
    }
  float cov00 = tmp[0][0]*T2[0][0] + tmp[0][1]*T2[0][1] + tmp[0][2]*T2[0][2];
  float cov01 = tmp[0][0]*T2[1][0] + tmp[0][1]*T2[1][1] + tmp[0][2]*T2[1][2];
  float cov11 = tmp[1][0]*T2[1][0] + tmp[1][1]*T2[1][1] + tmp[1][2]*T2[1][2];

  float a = cov00 + 0.3f, b = cov01, c = cov11 + 0.3f;
  float det = a*c - b*b;
  bool valid = (tz > 0.2f) && (det > 0.f);
  float inv = 1.0f / ((det > 0.f) ? det : 1.0f);
  float ca = c*inv, cb = -b*inv, cc = a*inv;

  float ph0 = projm[0]*mx  + projm[1]*my  + projm[2]*mz  + projm[3];
  float ph1 = projm[4]*mx  + projm[5]*my  + projm[6]*mz  + projm[7];
  float ph3 = projm[12]*mx + projm[13]*my + projm[14]*mz + projm[15];
  float pw  = 1.0f / (ph3 + 1e-7f);
  float px  = ((ph0*pw + 1.f)*(float)IMG - 1.f)*0.5f;
  float py  = ((ph1*pw + 1.f)*(float)IMG - 1.f)*0.5f;

  float dxc = mx - campos[0], dyc = my - campos[1], dzc = mz - campos[2];
  float dn = rsqrtf(dxc*dxc + dyc*dyc + dzc*dzc);
  float x = dxc*dn, y = dyc*dn, z = dzc*dn;
  float xx=x*x, yy=y*y, zz=z*z, xy=x*y, yz=y*z, xz=x*z;
  const float C0 = 0.28209479177387814f;
  const float C1 = 0.4886025119029199f;
  float bs[16];
  bs[0]  = C0;
  bs[1]  = -C1*y;  bs[2] = C1*z;  bs[3] = -C1*x;
  bs[4]  = 1.0925484305920792f*xy;
  bs[5]  = -1.0925484305920792f*yz;
  bs[6]  = 0.31539156525252005f*(2.f*zz - xx - yy);
  bs[7]  = -1.0925484305920792f*xz;
  bs[8]  = 0.5462742152960396f*(xx - yy);
  bs[9]  = -0.5900435899266435f*y*(3.f*xx - yy);
  bs[10] = 2.890611442640554f*xy*z;
  bs[11] = -0.4570457994644658f*y*(4.f*zz - xx - yy);
  bs[12] = 0.3731763325901154f*z*(2.f*zz - 3.f*xx - 3.f*yy);
  bs[13] = -0.4570457994644658f*x*(4.f*zz - xx - yy);
  bs[14] = 1.445305721320277f*z*(xx - yy);
  bs[15] = -0.5900435899266435f*x*(xx - 3.f*yy);
  float rgb[3];
  #pragma unroll
  for (int cch = 0; cch < 3; ++cch) {
    float s = bs[0]*p[3 + cch];
    #pragma unroll
    for (int k = 1; k < 16; ++k) s += bs[k]*p[6 + cch*15 + (k-1)];
    rgb[cch] = fmaxf(s + 0.5f, 0.f);
  }

  float opac = 1.0f / (1.0f + __expf(-p[51]));

  float g0 = -0.5f*(ca*px*px + cc*py*py) - cb*px*py;
  float g1 = ca*px + cb*py;
  float g2 = cc*py + cb*px;
  float g3 = -0.5f*ca;
  float g4 = -0.5f*cc;
  float g5 = -cb;

  float* rec = recs + g*REC_DW;
  rec[0]=g0; rec[1]=g1; rec[2]=g2;  rec[3]=g3;
  rec[4]=g4; rec[5]=g5; rec[6]=valid ? opac : 0.f; rec[7]=tz;
  rec[8]=rgb[0]; rec[9]=rgb[1]; rec[10]=rgb[2]; rec[11]=0.f;
}

// ---------------------------------------------------------------------------
// Kernel 2: single-block LDS bitonic sort on depth tz + gather into order.
// ---------------------------------------------------------------------------
__global__ __launch_bounds__(1024) void gs_sort_gather(
    const float* __restrict__ recs, float* __restrict__ sorted)
{
  __shared__ float key[NG];
  __shared__ int   val[NG];
  int tid = threadIdx.x;
  for (int i = tid; i < NG; i += 1024) { key[i] = recs[i*REC_DW + 7]; val[i] = i; }
  __syncthreads();
  for (int k = 2; k <= NG; k <<= 1) {
    for (int j = k >> 1; j > 0; j >>= 1) {
      for (int i = tid; i < NG; i += 1024) {
        int ixj = i ^ j;
        if (ixj > i) {
          bool up = ((i & k) == 0);
          float ki = key[i], kj = key[ixj];
          if ((ki > kj) == up) {
            key[i] = kj; key[ixj] = ki;
            int tv = val[i]; val[i] = val[ixj]; val[ixj] = tv;
          }
        }
      }
      __syncthreads();
    }
  }
  for (int i = tid; i < NG; i += 1024) {
    int s = val[i];
    const float4* src = (const float4*)(recs + (size_t)s*REC_DW);
    float4* dst = (float4*)(sorted + (size_t)i*REC_DW);
    dst[0] = src[0]; dst[1] = src[1]; dst[2] = src[2];
  }
}

// ---------------------------------------------------------------------------
// Kernel 3: WMMA blend.
//  - TDM DMAs the full 96KB sorted-record array into LDS once per workgroup
//    (one tensor_load_to_lds, TENSORcnt-tracked), removing the 8x-redundant
//    per-wave VMEM traffic; inner loop reads via ds_load_b128 broadcasts.
//  - Per 16-pixel x 16-gaussian tile: power = two chained
//    V_WMMA_F32_16X16X4_F32 over features [1,x,y,x^2 | y^2,xy,0,0].
//  - Front-to-back transmittance: width-16 cross-lane cumprod scan.
// ---------------------------------------------------------------------------
__global__ __launch_bounds__(256) void gs_blend(
    const float* __restrict__ sorted,
    const float* __restrict__ bg,
    float* __restrict__ out)
{
  extern __shared__ float smem[];   // NG*REC_DW floats, dynamic LDS (offset 0)

  // --- stage records into LDS with the Tensor Data Mover (wave 0 issues) ---
  if (threadIdx.x < 32) {
    unsigned long long ga = (unsigned long long)(const void*)sorted;
    const u32 ndw = NG * REC_DW;                 // 24576 dwords, fits 16-bit tile_dim0
    v4u g0;
    g0[0] = 1u;                                  // count=1, user-mode descriptor
    g0[1] = 0u;                                  // lds_addr = 0 (dynamic LDS base)
    g0[2] = (u32)(ga & 0xFFFFFFFFull);           // global_addr[31:0]
    g0[3] = (u32)((ga >> 32) & 0x1FFFFFFull)     // global_addr[56:32]
          | (2u << 30);                          // type = 2 ("image")
    v8i g1;
    g1[0] = (int)(2u << 16);                     // wg_mask=0, data_size=2 (4B), no pad/iter
    g1[1] = (int)((ndw & 0xFFFFu) << 16);        // tensor_dim0[15:0]  @ bits 63:48
    g1[2] = (int)((ndw >> 16) | (1u << 16));     // tensor_dim0[31:16] | tensor_dim1=1
    g1[3] = (int)((ndw & 0xFFFFu) << 16);        // tensor_dim1 hi=0 | tile_dim0 @ 127:112
    g1[4] = 1;                                   // tile_dim1=1, tile_dim2=0
    g1[5] = (int)ndw;                            // tensor_dim0_stride[31:0]
    g1[6] = 0;                                   // stride hi / dim1_stride lo
    g1[7] = 0;
    v4i gz4  = {0, 0, 0, 0};                     // groups 2/3 unused (<=2D tensor)
    v8i gz8  = {0, 0, 0, 0, 0, 0, 0, 0};
    __builtin_amdgcn_tensor_load_to_lds(g0, g1, gz4, gz4, gz8, 0);
    __builtin_amdgcn_s_wait_tensorcnt(0);
  }
  __syncthreads();

  const int lane = threadIdx.x & 31;
  const int hf   = lane >> 4;        // half of the wave
  const int sub  = lane & 15;        // lane within 16-lane group
  const int wave = blockIdx.x * (blockDim.x >> 5) + (threadIdx.x >> 5);
  const int base = wave * 16;        // first pixel of this wave's tile

  // A-matrix pixel features: lane holds pixel row M = sub for K = 2*hf, 2*hf+1
  const int pix = base + sub;
  const float fxp = (float)(pix & (IMG - 1));
  const float fyp = (float)(pix >> 7);
  float f[6];
  f[0]=1.f; f[1]=fxp; f[2]=fyp; f[3]=fxp*fxp; f[4]=fyp*fyp; f[5]=fxp*fyp;
  const int ks = hf * 2;
  v2f a1; a1.x = f[ks];            a1.y = f[ks+1];
  v2f a2; a2.x = hf ? 0.f : f[4];  a2.y = hf ? 0.f : f[5];

  float T[8], accR[8], accG[8], accB[8];
  #pragma unroll
  for (int r = 0; r < 8; ++r) { T[r]=1.f; accR[r]=0.f; accG[r]=0.f; accB[r]=0.f; }

  for (int chunk = 0; chunk < NG/16; ++chunk) {
    const float4* rp = (const float4*)(smem + (size_t)(chunk*16 + sub)*REC_DW);
    float4 r0 = rp[0];   // g0..g3
    float4 r1 = rp[1];   // g4, g5, opac', tz
    float4 r2 = rp[2];   // rgb
    float gb[4] = { r0.x, r0.y, r0.z, r0.w };
    v2f b1; b1.x = gb[ks];            b1.y = gb[ks+1];
    v2f b2; b2.x = hf ? 0.f : r1.x;   b2.y = hf ? 0.f : r1.y;
    float op = r1.z;

    v8f pw = {};
    pw = __builtin_amdgcn_wmma_f32_16x16x4_f32(false, a1, false, b1,
                                               (short)0, pw, false, false);
    pw = __builtin_amdgcn_wmma_f32_16x16x4_f32(false, a2, false, b2,
                                               (short)0, pw, false, false);

    #pragma unroll
    for (int r = 0; r < 8; ++r) {
      float power = pw[r];
      float al = fminf(0.99f, op * __expf(power));
      al = (power <= 0.f && al >= (1.0f/255.0f)) ? al : 0.f;
      float v = 1.f - al;
      // inclusive cumprod over the 16 gaussians (depth order = lane order)
      #pragma unroll
      for (int d = 1; d < 16; d <<= 1) {
        float t = __shfl_up(v, (unsigned)d, 16);
        if (sub >= d) v *= t;
      }
      float excl = __shfl_up(v, 1u, 16);
      if (sub == 0) excl = 1.f;
      float w = al * (T[r] * excl);
      accR[r] += w * r2.x;
      accG[r] += w * r2.y;
      accB[r] += w * r2.z;
      T[r] *= __shfl(v, 15, 16);   // chunk-total transmittance, broadcast
    }
  }

  // reduce color over the 16 gaussian lanes
  #pragma unroll
  for (int r = 0; r < 8; ++r) {
    #pragma unroll
    for (int d = 8; d >= 1; d >>= 1) {
      accR[r] += __shfl_xor(accR[r], d, 16);
      accG[r] += __shfl_xor(accG[r], d, 16);
      accB[r] += __shfl_xor(accB[r], d, 16);
    }
  }

  const float bg0 = bg[0], bg1 = bg[1], bg2 = bg[2];
  #pragma unroll
  for (int r = 0; r < 8; ++r) {
    if (sub == r) {
      int pixel = base + r + 8*hf;
      out[0*NPIX + pixel] = accR[r] + T[r]*bg0;
      out[1*NPIX + pixel] = accG[r] + T[r]*bg1;
      out[2*NPIX + pixel] = accB[r] + T[r]*bg2;
    }
  }
}

// ---------------------------------------------------------------------------
extern "C" void kernel_launch(void* const* d_in, const int* in_sizes, int n_in,
                              void* d_out, int out_size, void* d_ws, size_t ws_size,
                              hipStream_t stream) {
  const float* pc     = (const float*)d_in[0];
  const float* viewm  = (const float*)d_in[1];
  const float* projm  = (const float*)d_in[2];
  const float* campos = (const float*)d_in[3];
  const float* bg     = (const float*)d_in[4];
  float* out = (float*)d_out;

  float* recs   = (float*)d_ws;              // NG*12 floats
  float* sorted = recs + (size_t)NG*REC_DW;  // NG*12 floats

  gs_preprocess<<<NG/256, 256, 0, stream>>>(pc, viewm, projm, campos, recs);
  gs_sort_gather<<<1, 1024, 0, stream>>>(recs, sorted);
  // 16384 pixels / 16 per wave = 1024 waves = 128 blocks x 8 waves
  gs_blend<<<128, 256, SMEM_BYTES, stream>>>(sorted, bg, out);
}